// PartitionPadding_4690104287618
// MI455X (gfx1250) — compile-verified
//
#include <hip/hip_runtime.h>
#include <stdint.h>

// Problem constants (from the reference)
#define BATCH          1024
#define ATOMS_PER_MOL  512
#define DIM            128
#define NROWS          (BATCH * ATOMS_PER_MOL)   // 524288 rows
#define ROW_BYTES      (DIM * 4)                 // 512 bytes per feature row

// Copy-kernel geometry: one wave moves one 512B row per async op
#define RPW             16                        // rows staged per wave
#define WAVES_PER_BLOCK 8
#define THREADS         (WAVES_PER_BLOCK * 32)    // 256
#define ROWS_PER_BLOCK  (WAVES_PER_BLOCK * RPW)   // 128 rows -> 64KB LDS/block

// ---------------------------------------------------------------------------
// Kernel 0: offsets[b] = first index i with molecule_indicator[i] >= b
// (exclusive prefix sum of counts, via binary search on the sorted indicator)
// ---------------------------------------------------------------------------
__global__ void pp_offsets_kernel(const int* __restrict__ ind,
                                  int* __restrict__ offsets, int n) {
    int b = blockIdx.x * blockDim.x + threadIdx.x;
    if (b >= BATCH) return;
    int lo = 0, hi = n;
    while (lo < hi) {
        int mid = (lo + hi) >> 1;
        if (ind[mid] < b) lo = mid + 1; else hi = mid;
    }
    offsets[b] = lo;
}

// ---------------------------------------------------------------------------
// Kernel 1: streaming scatter through LDS using CDNA5 async global<->LDS DMA.
// Each wave owns a private RPW*512B LDS slice (no barriers needed):
//   A) issue RPW global_load_async_to_lds_b128  (ASYNCcnt deep-pipelined)
//   B) compute scattered destination rows on the SCALAR path while the
//      async loads are in flight (readfirstlane -> uniform s_loads + SALU)
//   C) s_wait_asynccnt 0   (async LDS writes may land out of order)
//   D) issue RPW global_store_async_from_lds_b128
// ---------------------------------------------------------------------------
__global__ __launch_bounds__(THREADS)
void pp_scatter_copy_kernel(const char* __restrict__ src,
                            const int*  __restrict__ ind,
                            const int*  __restrict__ offsets,
                            char* __restrict__ dst) {
    __shared__ char lds_buf[WAVES_PER_BLOCK * RPW * ROW_BYTES];

    const int tid  = (int)threadIdx.x;
    const int wave = tid >> 5;
    const int lane = tid & 31;
    const int row0 = ((int)blockIdx.x * WAVES_PER_BLOCK + wave) * RPW;
    const unsigned lane_b = (unsigned)lane * 16u;

    // Low 32 bits of a generic LDS pointer == LDS byte offset for the wave.
    const unsigned lbase =
        (unsigned)(uintptr_t)(&lds_buf[wave * RPW * ROW_BYTES]);

    // ---- A: stage RPW rows, global -> LDS, async (ASYNCcnt-tracked) ----
    #pragma unroll
    for (int r = 0; r < RPW; ++r) {
        unsigned voff = (unsigned)(row0 + r) * (unsigned)ROW_BYTES + lane_b;
        unsigned loff = lbase + (unsigned)r * (unsigned)ROW_BYTES + lane_b;
        asm volatile("global_load_async_to_lds_b128 %0, %1, %2"
                     :
                     : "v"(loff), "v"(voff), "s"(src)
                     : "memory");
    }

    // ---- B: destination rows on the scalar path, overlapped with A ----
    // row0 is wave-uniform; make that provable so ind[]/offsets[] lower to
    // uniform (SMEM) loads and the address math stays in SALU.
    const int row0_s = __builtin_amdgcn_readfirstlane(row0);
    int dst_row[RPW];
    #pragma unroll
    for (int r = 0; r < RPW; ++r) {
        int row = row0_s + r;
        int mol = ind[row];                          // uniform -> s_load
        dst_row[r] = mol * ATOMS_PER_MOL + (row - offsets[mol]);
    }

    // ---- C: drain async loads before reading their LDS destinations ----
    asm volatile("s_wait_asynccnt 0" ::: "memory");

    // ---- D: scatter RPW rows, LDS -> global, async ----
    #pragma unroll
    for (int r = 0; r < RPW; ++r) {
        unsigned voff = (unsigned)dst_row[r] * (unsigned)ROW_BYTES + lane_b;
        unsigned loff = lbase + (unsigned)r * (unsigned)ROW_BYTES + lane_b;
        asm volatile("global_store_async_from_lds_b128 %0, %1, %2"
                     :
                     : "v"(voff), "v"(loff), "s"(dst)
                     : "memory");
    }

    // Be explicit: all async stores retired before wave end.
    asm volatile("s_wait_asynccnt 0" ::: "memory");
}

// ---------------------------------------------------------------------------
// Host-side launcher
// ---------------------------------------------------------------------------
extern "C" void kernel_launch(void* const* d_in, const int* in_sizes, int n_in,
                              void* d_out, int out_size, void* d_ws, size_t ws_size,
                              hipStream_t stream) {
    (void)in_sizes; (void)n_in; (void)out_size; (void)ws_size;

    const float* feat = (const float*)d_in[0];           // (N, 128) f32
    const int*   ind  = (const int*)d_in[1];             // (N,) sorted int32
    int*         offs = (int*)d_ws;                      // 1024 ints scratch

    pp_offsets_kernel<<<(BATCH + 255) / 256, 256, 0, stream>>>(ind, offs, NROWS);

    const int blocks = NROWS / ROWS_PER_BLOCK;           // 4096
    pp_scatter_copy_kernel<<<blocks, THREADS, 0, stream>>>(
        (const char*)feat, ind, offs, (char*)d_out);
}